// LearnableQueryAttentionPool1d_7189775253652
// MI455X (gfx1250) — compile-verified
//
#include <hip/hip_runtime.h>
#include <hip/hip_bf16.h>

#define B_ 16
#define N_ 4096
#define D_ 1024
#define H_ 16
#define HD_ 64
#define NSPLIT 16
#define NCHUNK (N_ / NSPLIT) /* 256 */

typedef __attribute__((ext_vector_type(2))) float v2f;
typedef __attribute__((ext_vector_type(8))) float v8f;

// ---------------------------------------------------------------------------
// Kernel 1a: q[e] = scale * (query . Wq[e,:])  — wave per output, coalesced.
// ---------------------------------------------------------------------------
__global__ __launch_bounds__(256) void qproj_kernel(
    const float* __restrict__ query, const float* __restrict__ Wq,
    float* __restrict__ q) {
  const int gw = (blockIdx.x * 256 + threadIdx.x) >> 5; // 0..1023
  const int lane = threadIdx.x & 31;
  const float* wr = Wq + (size_t)gw * D_;
  float s = 0.f;
  for (int d = lane; d < D_; d += 32) s += query[d] * wr[d];
  for (int off = 16; off > 0; off >>= 1) s += __shfl_down(s, off, 32);
  if (lane == 0) q[gw] = s * 0.125f; // 1/sqrt(64)
}

// ---------------------------------------------------------------------------
// Kernel 1b: qk[h,d] = sum_e q[h*64+e] * Wk[h*64+e, d] — thread per output.
// ---------------------------------------------------------------------------
__global__ __launch_bounds__(256) void qk_kernel(
    const float* __restrict__ q, const float* __restrict__ Wk,
    float* __restrict__ qk) {
  const int flat = blockIdx.x * 256 + threadIdx.x; // H_*D_ = 16384
  const int h = flat >> 10;
  const int d = flat & (D_ - 1);
  float s = 0.f;
  for (int e = 0; e < HD_; ++e)
    s += q[h * HD_ + e] * Wk[(size_t)(h * HD_ + e) * D_ + d];
  qk[flat] = s;
}

// ---------------------------------------------------------------------------
// Kernel 2: scores[b,h,n] = qk[h,:] . x[b,n,:]   (GEMM, M=16 heads)
// One wave per 16-token tile, V_WMMA_F32_16X16X4_F32, K-loop over D.
// ---------------------------------------------------------------------------
__global__ __launch_bounds__(32) void scores_kernel(
    const float* __restrict__ x, const float* __restrict__ qk,
    float* __restrict__ scores) {
  const int lane = threadIdx.x;
  const int nlo = lane & 15;
  const int hi = lane >> 4;
  const int tile = blockIdx.x;       // B_ * (N_/16)
  const int b = tile >> 8;           // 256 tiles per batch
  const int nbase = (tile & 255) << 4;
  const float* arow = qk + (size_t)nlo * D_;                   // A: M=head
  const float* brow = x + (size_t)(b * N_ + nbase + nlo) * D_; // B: N=token
  v8f c = {};
#pragma unroll 8
  for (int k = 0; k < D_; k += 4) {
    v2f a = *(const v2f*)(arow + k + 2 * hi);
    v2f bb = *(const v2f*)(brow + k + 2 * hi);
    c = __builtin_amdgcn_wmma_f32_16x16x4_f32(false, a, false, bb, (short)0, c,
                                              false, false);
  }
#pragma unroll
  for (int r = 0; r < 8; ++r) {
    int h = r + 8 * hi;
    scores[(size_t)(b * H_ + h) * N_ + nbase + nlo] = c[r];
  }
}

// ---------------------------------------------------------------------------
// Kernel 3: row softmax over n for each (b,h)
// ---------------------------------------------------------------------------
__global__ void softmax_kernel(float* __restrict__ scores) {
  __shared__ float red[256];
  float* row = scores + (size_t)blockIdx.x * N_;
  const int tid = threadIdx.x;
  float m = -1e30f;
  for (int i = tid; i < N_; i += 256) m = fmaxf(m, row[i]);
  red[tid] = m;
  __syncthreads();
  for (int s = 128; s > 0; s >>= 1) {
    if (tid < s) red[tid] = fmaxf(red[tid], red[tid + s]);
    __syncthreads();
  }
  m = red[0];
  __syncthreads();
  float sum = 0.f;
  for (int i = tid; i < N_; i += 256) {
    float e = __expf(row[i] - m);
    row[i] = e;
    sum += e;
  }
  red[tid] = sum;
  __syncthreads();
  for (int s = 128; s > 0; s >>= 1) {
    if (tid < s) red[tid] += red[tid + s];
    __syncthreads();
  }
  float inv = 1.0f / red[0];
  for (int i = tid; i < N_; i += 256) row[i] *= inv;
}

// ---------------------------------------------------------------------------
// Kernel 4: split-K partial ctx[b,h,d] = sum_n attn[b,h,n]*x[b,n,d]
// GEMM M=16 heads, N-dim = d tile of 16, K = sequence chunk of 256.
// ---------------------------------------------------------------------------
__global__ __launch_bounds__(32) void ctx_kernel(
    const float* __restrict__ x, const float* __restrict__ attn,
    float* __restrict__ pctx) {
  const int lane = threadIdx.x;
  const int nlo = lane & 15;
  const int hi = lane >> 4;
  int id = blockIdx.x;               // B_ * 64 * NSPLIT
  const int split = id & (NSPLIT - 1); id >>= 4;
  const int dt = id & 63;             id >>= 6;
  const int b = id;
  const int dbase = dt << 4;
  const int n0 = split * NCHUNK;
  const float* arow = attn + (size_t)(b * H_ + nlo) * N_ + n0; // A: M=head
  const float* xb = x + (size_t)b * N_ * D_;
  v8f c = {};
#pragma unroll 4
  for (int k = 0; k < NCHUNK; k += 4) {
    v2f a = *(const v2f*)(arow + k + 2 * hi);
    v2f bb;
    bb.x = xb[(size_t)(n0 + k + 2 * hi) * D_ + dbase + nlo];
    bb.y = xb[(size_t)(n0 + k + 2 * hi + 1) * D_ + dbase + nlo];
    c = __builtin_amdgcn_wmma_f32_16x16x4_f32(false, a, false, bb, (short)0, c,
                                              false, false);
  }
#pragma unroll
  for (int r = 0; r < 8; ++r) {
    int h = r + 8 * hi;
    pctx[(((size_t)split * B_ + b) * H_ + h) * D_ + dbase + nlo] = c[r];
  }
}

// ---------------------------------------------------------------------------
// Kernel 5: reduce split-K partials
// ---------------------------------------------------------------------------
__global__ void ctx_reduce_kernel(const float* __restrict__ pctx,
                                  float* __restrict__ ctx) {
  int i = blockIdx.x * blockDim.x + threadIdx.x; // B_*H_*D_
  float s = 0.f;
  for (int sp = 0; sp < NSPLIT; ++sp)
    s += pctx[(size_t)sp * (B_ * H_ * D_) + i];
  ctx[i] = s;
}

// ---------------------------------------------------------------------------
// Kernel 6: o[b, h*64+e] = Wv[h*64+e,:] . ctx[b,h,:]
// GEMM M=16 batches; 8 waves per block split K, LDS reduction.
// ---------------------------------------------------------------------------
__global__ __launch_bounds__(256) void vproj_kernel(
    const float* __restrict__ Wv, const float* __restrict__ ctx,
    float* __restrict__ o) {
  __shared__ float red[8 * 256];
  const int tid = threadIdx.x;
  const int wave = tid >> 5;  // K-split 0..7
  const int lane = tid & 31;
  const int nlo = lane & 15;
  const int hi = lane >> 4;
  const int h = blockIdx.x >> 2;
  const int sub = blockIdx.x & 3;
  const float* arow = ctx + (size_t)(nlo * H_ + h) * D_;            // A: M=batch
  const float* brow = Wv + (size_t)(h * HD_ + sub * 16 + nlo) * D_; // B: N=out
  const int k0 = wave * (D_ / 8);
  v8f c = {};
#pragma unroll 8
  for (int k = k0; k < k0 + D_ / 8; k += 4) {
    v2f a = *(const v2f*)(arow + k + 2 * hi);
    v2f bb = *(const v2f*)(brow + k + 2 * hi);
    c = __builtin_amdgcn_wmma_f32_16x16x4_f32(false, a, false, bb, (short)0, c,
                                              false, false);
  }
#pragma unroll
  for (int r = 0; r < 8; ++r) red[wave * 256 + lane * 8 + r] = c[r];
  __syncthreads();
  float s = 0.f;
#pragma unroll
  for (int w = 0; w < 8; ++w) s += red[w * 256 + tid];
  const int l = tid >> 3;
  const int r = tid & 7;
  const int bidx = r + 8 * (l >> 4);
  const int col = l & 15;
  o[(size_t)bidx * D_ + h * HD_ + sub * 16 + col] = s;
}

// ---------------------------------------------------------------------------
// Kernel 7: out[b,j] = o[b,:] . Wc[j,:]  — same split-K structure.
// ---------------------------------------------------------------------------
__global__ __launch_bounds__(256) void outproj_kernel(
    const float* __restrict__ Wc, const float* __restrict__ o,
    float* __restrict__ out) {
  __shared__ float red[8 * 256];
  const int tid = threadIdx.x;
  const int wave = tid >> 5;
  const int lane = tid & 31;
  const int nlo = lane & 15;
  const int hi = lane >> 4;
  const int jbase = blockIdx.x * 16; // D_/16 blocks
  const float* arow = o + (size_t)nlo * D_;            // A: M=batch
  const float* brow = Wc + (size_t)(jbase + nlo) * D_; // B: N=out col
  const int k0 = wave * (D_ / 8);
  v8f c = {};
#pragma unroll 8
  for (int k = k0; k < k0 + D_ / 8; k += 4) {
    v2f a = *(const v2f*)(arow + k + 2 * hi);
    v2f bb = *(const v2f*)(brow + k + 2 * hi);
    c = __builtin_amdgcn_wmma_f32_16x16x4_f32(false, a, false, bb, (short)0, c,
                                              false, false);
  }
#pragma unroll
  for (int r = 0; r < 8; ++r) red[wave * 256 + lane * 8 + r] = c[r];
  __syncthreads();
  float s = 0.f;
#pragma unroll
  for (int w = 0; w < 8; ++w) s += red[w * 256 + tid];
  const int l = tid >> 3;
  const int r = tid & 7;
  const int bidx = r + 8 * (l >> 4);
  const int col = l & 15;
  out[(size_t)bidx * D_ + jbase + col] = s;
}

extern "C" void kernel_launch(void* const* d_in, const int* in_sizes, int n_in,
                              void* d_out, int out_size, void* d_ws,
                              size_t ws_size, hipStream_t stream) {
  (void)in_sizes; (void)n_in; (void)out_size; (void)ws_size;
  const float* x     = (const float*)d_in[0];
  const float* query = (const float*)d_in[1];
  const float* Wq    = (const float*)d_in[2];
  const float* Wk    = (const float*)d_in[3];
  const float* Wv    = (const float*)d_in[4];
  const float* Wc    = (const float*)d_in[5];
  float* out = (float*)d_out;

  float* ws     = (float*)d_ws;
  float* q      = ws;                                   // D_
  float* qk     = q + D_;                               // H*D
  float* scores = qk + H_ * D_;                         // B*H*N (softmax in place)
  float* pctx   = scores + (size_t)B_ * H_ * N_;        // NSPLIT*B*H*D
  float* ctx    = pctx + (size_t)NSPLIT * B_ * H_ * D_; // B*H*D
  float* o      = ctx + (size_t)B_ * H_ * D_;           // B*D

  hipLaunchKernelGGL(qproj_kernel, dim3(D_ / 8), dim3(256), 0, stream,
                     query, Wq, q);
  hipLaunchKernelGGL(qk_kernel, dim3((H_ * D_) / 256), dim3(256), 0, stream,
                     q, Wk, qk);
  hipLaunchKernelGGL(scores_kernel, dim3(B_ * (N_ / 16)), dim3(32), 0, stream,
                     x, qk, scores);
  hipLaunchKernelGGL(softmax_kernel, dim3(B_ * H_), dim3(256), 0, stream,
                     scores);
  hipLaunchKernelGGL(ctx_kernel, dim3(B_ * (D_ / 16) * NSPLIT), dim3(32), 0,
                     stream, x, scores, pctx);
  hipLaunchKernelGGL(ctx_reduce_kernel, dim3((B_ * H_ * D_) / 256), dim3(256),
                     0, stream, pctx, ctx);
  hipLaunchKernelGGL(vproj_kernel, dim3(H_ * (HD_ / 16)), dim3(256), 0, stream,
                     Wv, ctx, o);
  hipLaunchKernelGGL(outproj_kernel, dim3(D_ / 16), dim3(256), 0, stream,
                     Wc, o, out);
}